// SignaxOnly_68633577390310
// MI455X (gfx1250) — compile-verified
//
#include <hip/hip_runtime.h>

typedef __attribute__((ext_vector_type(2))) float v2f;
typedef __attribute__((ext_vector_type(8))) float v8f;

#define LPATH   65536
#define NINC    (LPATH - 1)        // 65535 increments
#define NCHUNK  1024
#define LSEG    64                 // 1024*64 = 65536 >= 65535
#define SIG_LEN 1554               // 6 + 36 + 216 + 1296

// padded per-level layout inside a slot (all level bases 16-aligned)
#define P1   0                     // 6  valid, pad to 16
#define P2   16                    // 36 valid, pad to 48
#define P3   64                    // 216 valid, pad to 224
#define P4   288                   // 1296 valid, ends 1584
#define SLOT 1600                  // slot stride in floats (16 tail pad)

// combine-kernel LDS map (floats)
#define ACC   0
#define BOF   1600
#define DUMP  3200                 // 32-float scatter dump
#define ZID   3232                 // guaranteed-zero word (16 floats)
#define LDSF  3248

// ---------------------------------------------------------------------------
// Phase 1: per-chunk serial signature fold (Horner form of S <- S (x) exp(dx)).
// One wave per chunk.  Small state (S1/S2/u2/v2/d) in padded LDS so all
// cross-index gathers are plain dynamic ds_loads with no EXEC masking;
// big state (S3: 7 rows/lane, S4: 42 f32/lane) statically indexed in VGPRs.
// ---------------------------------------------------------------------------
__global__ __launch_bounds__(32) void sig_phase1(const float* __restrict__ x,
                                                 float* __restrict__ part)
{
    const int lane = threadIdx.x;
    const int c  = blockIdx.x;
    const int t0 = c * LSEG;
    const int t1 = min(t0 + LSEG, NINC);

    __shared__ float s1lds[16];
    __shared__ float s2lds[64];
    __shared__ float dlds[16];
    __shared__ __align__(16) float uvlds[128];  // interleaved (u2, v2) per ij

    s1lds[lane % 16] = 0.f;                     // lanes 0..15 + 16..31 both zero it
    dlds[lane % 16]  = 0.f;
    s2lds[lane] = 0.f; s2lds[lane + 32] = 0.f;

    // owned rows r = lane + 32*q (garbage rows r>=216 computed but never stored)
    int ij6[7], kk[7];
#pragma unroll
    for (int q = 0; q < 7; ++q) {
        int r = lane + 32 * q;
        ij6[q] = r / 6;
        kk[q]  = r % 6;
    }

    float s3o[7];
    float s4o[42];
#pragma unroll
    for (int q = 0; q < 7; ++q) s3o[q] = 0.f;
#pragma unroll
    for (int i = 0; i < 42; ++i) s4o[i] = 0.f;

    float xp[6];
#pragma unroll
    for (int j = 0; j < 6; ++j) xp[j] = x[(size_t)t0 * 6 + j];

    __syncthreads();

    for (int t = t0; t < t1; ++t) {
        float d[6];
#pragma unroll
        for (int j = 0; j < 6; ++j) {
            float xn = x[(size_t)(t + 1) * 6 + j];
            d[j] = xn - xp[j];
            xp[j] = xn;
        }
        if (t + 2 < LPATH) __builtin_prefetch(&x[(size_t)(t + 2) * 6], 0, 1);

        if (lane == 0) {
#pragma unroll
            for (int j = 0; j < 6; ++j) dlds[j] = d[j];
        }
        __syncthreads();

        // u2 = S2 + (S1 + dx/4)(dx/3);  v2 = S2 + (S1 + dx/3)(dx/2);
        // S2 <- S2 + (S1 + dx/2)(dx)    (reads OLD S1/S2; pads compute garbage)
#pragma unroll
        for (int rep = 0; rep < 2; ++rep) {
            int idx = lane + 32 * rep;          // 0..63, valid < 36
            int i = idx / 6, j = idx % 6;
            float s1v = s1lds[i];
            float di  = dlds[i];
            float dj  = dlds[j];
            float s2v = s2lds[idx];
            float u2  = fmaf(fmaf(0.25f, di, s1v), dj * (1.f / 3.f), s2v);
            float v2  = fmaf(fmaf(1.f / 3.f, di, s1v), 0.5f * dj, s2v);
            uvlds[2 * idx]     = u2;
            uvlds[2 * idx + 1] = v2;
            s2lds[idx] = fmaf(fmaf(0.5f, di, s1v), dj, s2v);
        }
        __syncthreads();

        // S4 <- S4 + (S3 + u2 (x) dx/2) (x) dx ;  S3 <- S3 + v2 (x) dx
#pragma unroll
        for (int q = 0; q < 7; ++q) {
            float2 uv = *(const float2*)&uvlds[2 * ij6[q]];
            float  dk = dlds[kk[q]];
            float  u3 = fmaf(uv.x, 0.5f * dk, s3o[q]);
#pragma unroll
            for (int l = 0; l < 6; ++l)
                s4o[q * 6 + l] = fmaf(u3, d[l], s4o[q * 6 + l]);
            s3o[q] = fmaf(uv.y, dk, s3o[q]);
        }
        if (lane < 6) s1lds[lane] += dlds[lane];
        __syncthreads();
    }

    // write this chunk's partial signature (padded layout, pads zeroed)
    float* slot = part + (size_t)c * SLOT;
    if (lane < 6)  slot[P1 + lane] = s1lds[lane];
    if (lane < 10) slot[P1 + 6 + lane] = 0.f;           // L1 pad
    for (int i = lane; i < 36; i += 32) slot[P2 + i] = s2lds[i];
    if (lane < 12) slot[P2 + 36 + lane] = 0.f;          // L2 pad
    if (lane < 8)  slot[P3 + 216 + lane] = 0.f;         // L3 pad
    if (lane < 16) slot[P4 + 1296 + lane] = 0.f;        // tail pad
#pragma unroll
    for (int q = 0; q < 7; ++q) {
        int r = lane + 32 * q;
        if (r < 216) {
            slot[P3 + r] = s3o[q];
#pragma unroll
            for (int l = 0; l < 6; ++l)
                slot[P4 + r * 6 + l] = s4o[q * 6 + l];
        }
    }
}

// ---------------------------------------------------------------------------
// Rank-1 outer product Avec (x) Bvec scatter-added into an LDS tensor via
// V_WMMA_F32_16X16X4_F32 (K padded to 1).  Branch-free:
//  - operands come from zero-padded regions; lanes 16..31 (K=2,3 half)
//    redirect to a zeroed word via address select,
//  - scatter uses ds_add_f32 with a dump-region redirect for OOB lanes.
// A layout: lanes 0-15 VGPR0 = A[m][k=0].  B: lanes 0-15 VGPR0 = B[k=0][n].
// D tile: row = 8*(lane>=16) + vgpr, col = lane%16.
// ---------------------------------------------------------------------------
template<int Na, int Nb>
__device__ __forceinline__ void rank1_add(float* lds, int aOff, int bOff,
                                          int dstOff, int lane)
{
    constexpr int MT = (Na + 15) / 16;
    constexpr int NT = (Nb + 15) / 16;
    const int  l15 = lane & 15;
    const bool lo  = lane < 16;
    const int  hi8 = (lane >> 4) << 3;

    float av[MT], bv[NT];
#pragma unroll
    for (int mt = 0; mt < MT; ++mt)
        av[mt] = lds[lo ? (aOff + mt * 16 + l15) : ZID];
#pragma unroll
    for (int nt = 0; nt < NT; ++nt)
        bv[nt] = lds[lo ? (bOff + nt * 16 + l15) : ZID];

#pragma unroll
    for (int mt = 0; mt < MT; ++mt) {
        v2f a; a.x = av[mt]; a.y = 0.f;
#pragma unroll
        for (int nt = 0; nt < NT; ++nt) {
            v2f b; b.x = bv[nt]; b.y = 0.f;
            v8f cz = {};
            v8f dd = __builtin_amdgcn_wmma_f32_16x16x4_f32(
                false, a, false, b, (short)0, cz, false, false);
#pragma unroll
            for (int e = 0; e < 8; ++e) {
                int row = mt * 16 + hi8 + e;
                int col = nt * 16 + l15;
                bool inb = (row < Na) & (col < Nb);
                int  addr = inb ? (dstOff + row * Nb + col) : (DUMP + lane);
                atomicAdd(&lds[addr], dd[e]);
            }
        }
    }
}

// Chen combine in place: acc <- acc (x) B.  Level order preserves reads of
// old acc1..acc3 before they are overwritten (same-wave LDS is in-order).
__device__ __forceinline__ void combine_inplace(float* lds, int lane)
{
    // level 4 cross terms -> acc4
    rank1_add<6, 216>(lds, ACC + P1, BOF + P3, ACC + P4, lane);  // A1 (x) B3
    rank1_add<36, 36>(lds, ACC + P2, BOF + P2, ACC + P4, lane);  // A2 (x) B2
    rank1_add<216, 6>(lds, ACC + P3, BOF + P1, ACC + P4, lane);  // A3 (x) B1
    // level 3 cross terms -> acc3
    rank1_add<6, 36>(lds, ACC + P1, BOF + P2, ACC + P3, lane);   // A1 (x) B2
    rank1_add<36, 6>(lds, ACC + P2, BOF + P1, ACC + P3, lane);   // A2 (x) B1
    // level 2 cross term -> acc2
    rank1_add<6, 6>(lds, ACC + P1, BOF + P1, ACC + P2, lane);    // A1 (x) B1
    __syncthreads();
    for (int i = lane; i < SLOT; i += 32) lds[ACC + i] += lds[BOF + i];
    __syncthreads();
}

// Fold 4 consecutive partial signatures into one (path order preserved).
// finalOut != 0: write packed 1554-float reference layout to `out`.
__global__ __launch_bounds__(32) void sig_combine(const float* __restrict__ in,
                                                  float* __restrict__ out,
                                                  int finalOut)
{
    __shared__ __align__(16) float lds[LDSF];
    const int lane = threadIdx.x;
    const int b = blockIdx.x;
    const size_t base = (size_t)b * 4;

    for (int i = lane; i < SLOT; i += 32) lds[ACC + i] = in[base * SLOT + i];
    lds[DUMP + lane] = 0.f;
    if (lane < 16) lds[ZID + lane] = 0.f;
    __syncthreads();
#pragma unroll 1
    for (int m = 1; m < 4; ++m) {
        for (int i = lane; i < SLOT; i += 32)
            lds[BOF + i] = in[(base + (size_t)m) * SLOT + i];
        __syncthreads();
        combine_inplace(lds, lane);
    }
    if (finalOut) {
        if (lane < 6) out[lane] = lds[P1 + lane];
        for (int i = lane; i < 36;   i += 32) out[6   + i] = lds[P2 + i];
        for (int i = lane; i < 216;  i += 32) out[42  + i] = lds[P3 + i];
        for (int i = lane; i < 1296; i += 32) out[258 + i] = lds[P4 + i];
    } else {
        for (int i = lane; i < SLOT; i += 32)
            out[(size_t)b * SLOT + i] = lds[ACC + i];
    }
}

// ---------------------------------------------------------------------------
extern "C" void kernel_launch(void* const* d_in, const int* in_sizes, int n_in,
                              void* d_out, int out_size, void* d_ws, size_t ws_size,
                              hipStream_t stream)
{
    const float* x = (const float*)d_in[0];
    float* out = (float*)d_out;
    float* R0 = (float*)d_ws;                       // 1024 slots
    float* R1 = R0 + (size_t)NCHUNK * SLOT;         // 256 slots
    (void)in_sizes; (void)n_in; (void)out_size; (void)ws_size;

    sig_phase1<<<NCHUNK, 32, 0, stream>>>(x, R0);
    // fold-4 tree: 1024 -> 256 -> 64 -> 16 -> 4 -> 1
    sig_combine<<<256, 32, 0, stream>>>(R0, R1, 0);
    sig_combine<<<64,  32, 0, stream>>>(R1, R0, 0);
    sig_combine<<<16,  32, 0, stream>>>(R0, R1, 0);
    sig_combine<<<4,   32, 0, stream>>>(R1, R0, 0);
    sig_combine<<<1,   32, 0, stream>>>(R0, out, 1);
}